// _MultiHeadAttention_71408126263357
// MI455X (gfx1250) — compile-verified
//
#include <hip/hip_runtime.h>
#include <hip/hip_bf16.h>
#include <math.h>

typedef float v2f __attribute__((ext_vector_type(2)));
typedef float v8f __attribute__((ext_vector_type(8)));

#define WMMA_F32(a_, b_, c_) \
  __builtin_amdgcn_wmma_f32_16x16x4_f32(false, (a_), false, (b_), (short)0, (c_), false, false)

// Problem constants: B=2, S=2048, D=768, H=12, DK=DV=64

// ---------------------------------------------------------------------------
// Tiled fp32-WMMA GEMM: out = A[MxK] * W[KxN] + bias[N]
// mode 0: row-major out[M][N]
// mode 1: scatter to [b][h][s][d]   (Q / V layout)
// mode 2: scatter to [b][h][d][s]   (K-transposed layout)
// ---------------------------------------------------------------------------
__device__ __forceinline__ void store_tile(v8f c, int m0, int n0, int tr, int tc,
                                           int lh, int ln, const float* __restrict__ bias,
                                           float* __restrict__ out, int N, int mode) {
  const int col = n0 + tc * 16 + ln;
  const float bv = bias[col];
#pragma unroll
  for (int g = 0; g < 8; ++g) {
    const int row = m0 + tr * 16 + g + 8 * lh;
    const float val = c[g] + bv;
    if (mode == 0) {
      out[(size_t)row * N + col] = val;
    } else {
      const int bb = row >> 11;      // row / 2048
      const int s  = row & 2047;
      const int hh = col >> 6;       // col / 64
      const int d  = col & 63;
      if (mode == 1)
        out[(((size_t)bb * 12 + hh) * 2048 + s) * 64 + d] = val;
      else
        out[(((size_t)bb * 12 + hh) * 64 + d) * 2048 + s] = val;
    }
  }
}

__global__ __launch_bounds__(256) void gemm_bias_64x64(
    const float* __restrict__ A, const float* __restrict__ W,
    const float* __restrict__ bias, float* __restrict__ out,
    int M, int N, int K, int mode) {
  __shared__ float As[64][68];
  __shared__ float Ws[64][68];

  const int tid = threadIdx.x;
  const int wv = tid >> 5;
  const int lane = tid & 31;
  const int lh = lane >> 4;   // lane-half
  const int ln = lane & 15;
  const int n0 = blockIdx.x * 64;
  const int m0 = blockIdx.y * 64;

  // Tiles 2w and 2w+1 always share the tile-row -> shared A fragments.
  const int t0 = wv * 2;
  const int tr  = t0 >> 2;
  const int tc0 = t0 & 3, tc1 = (t0 + 1) & 3;

  v8f acc0 = {0.f, 0.f, 0.f, 0.f, 0.f, 0.f, 0.f, 0.f};
  v8f acc1 = {0.f, 0.f, 0.f, 0.f, 0.f, 0.f, 0.f, 0.f};

  for (int kb = 0; kb < K; kb += 64) {
    __syncthreads();
    // ---- stage 64x64 A and W tiles via float4 (b128) -----------------------
#pragma unroll
    for (int i = 0; i < 4; ++i) {
      const int idx = tid + i * 256;       // float4 index, 0..1023
      const int r = idx >> 4;              // 16 float4 per 64-float row
      const int c = (idx & 15) * 4;
      *(float4*)&As[r][c] = *(const float4*)&A[(size_t)(m0 + r) * K + kb + c];
      *(float4*)&Ws[r][c] = *(const float4*)&W[(size_t)(kb + r) * N + n0 + c];
    }
    __syncthreads();

    // ---- preload all fragments for this k-block, then dense wmma chain -----
    v2f afr[16], bfr0[16], bfr1[16];
#pragma unroll
    for (int k4 = 0; k4 < 16; ++k4) {
      const int kk = k4 * 4 + 2 * lh;
      afr[k4].x  = As[tr * 16 + ln][kk];
      afr[k4].y  = As[tr * 16 + ln][kk + 1];
      bfr0[k4].x = Ws[kk][tc0 * 16 + ln];
      bfr0[k4].y = Ws[kk + 1][tc0 * 16 + ln];
      bfr1[k4].x = Ws[kk][tc1 * 16 + ln];
      bfr1[k4].y = Ws[kk + 1][tc1 * 16 + ln];
    }
#pragma unroll
    for (int k4 = 0; k4 < 16; ++k4) {
      acc0 = WMMA_F32(afr[k4], bfr0[k4], acc0);
      acc1 = WMMA_F32(afr[k4], bfr1[k4], acc1);
    }
  }
  store_tile(acc0, m0, n0, tr, tc0, lh, ln, bias, out, N, mode);
  store_tile(acc1, m0, n0, tr, tc1, lh, ln, bias, out, N, mode);
}

// ---------------------------------------------------------------------------
// Fused attention: block = one (b, h, 16-query-row tile); 8 waves, 256 thr.
// Q: [b][h][s][64], KT: [b][h][64][s], V: [b][h][s][64]
// attn: [b][h][q][k], ctx: row-major [4096 x 768]
// ---------------------------------------------------------------------------
__global__ __launch_bounds__(256) void attention_kernel(
    const float* __restrict__ Q, const float* __restrict__ KT,
    const float* __restrict__ V, const float* __restrict__ mask,
    float* __restrict__ attn, float* __restrict__ ctx) {
  __shared__ float qtile[16][68];
  __shared__ float wstat[8][16][2];
  __shared__ float rowstat[16][2];
  __shared__ float pstage[8][16][20];
  __shared__ float ctxpart[8][16][68];

  const int tid = threadIdx.x;
  const int wv = tid >> 5;
  const int lane = tid & 31;
  const int lh = lane >> 4;
  const int ln = lane & 15;

  const int bid = blockIdx.x;
  const int qt = bid & 127;                 // S/16 = 128 query tiles
  const int h = (bid >> 7) % 12;
  const int b = bid / (128 * 12);
  const int q0 = qt * 16;
  const size_t bh = (size_t)b * 12 + h;

  // stage Q tile (16 x 64) with float4
  {
    const float* qsrc = Q + (bh * 2048 + q0) * 64;
    const int idx = tid;                    // 256 float4 = 1024 floats
    const int r = idx >> 4;
    const int c = (idx & 15) * 4;
    *(float4*)&qtile[r][c] = *(const float4*)&qsrc[r * 64 + c];
  }
  __syncthreads();

  const float scale = 0.125f;               // 1/sqrt(64)
  float m_run[8], l_run[8];
#pragma unroll
  for (int g = 0; g < 8; ++g) { m_run[g] = -1e30f; l_run[g] = 0.f; }

  v8f ct[16];
  const float* ktbase = KT + bh * 64 * 2048;
  const float* mbase  = mask + (bh * 2048 + q0) * 2048;

  // ---- phase 1: raw scores + online softmax stats --------------------------
#pragma unroll
  for (int t = 0; t < 16; ++t) {
    const int n0 = (t * 8 + wv) * 16;

    // preload all 16 KT fragments (global) before the wmma chain
    v2f bfr[16];
#pragma unroll
    for (int k4 = 0; k4 < 16; ++k4) {
      const float* kp = ktbase + (size_t)(k4 * 4 + 2 * lh) * 2048 + n0 + ln;
      bfr[k4].x = kp[0];
      bfr[k4].y = kp[2048];
    }
    v8f c = {0.f, 0.f, 0.f, 0.f, 0.f, 0.f, 0.f, 0.f};
#pragma unroll
    for (int k4 = 0; k4 < 16; ++k4) {
      v2f a;
      a.x = qtile[ln][k4 * 4 + 2 * lh];
      a.y = qtile[ln][k4 * 4 + 2 * lh + 1];
      c = WMMA_F32(a, bfr[k4], c);
    }
#pragma unroll
    for (int g = 0; g < 8; ++g)
      c[g] = c[g] * scale + mbase[(size_t)(g + 8 * lh) * 2048 + n0 + ln];
    ct[t] = c;
#pragma unroll
    for (int g = 0; g < 8; ++g) {
      float tm = c[g];
      tm = fmaxf(tm, __shfl_xor(tm, 1, 32));
      tm = fmaxf(tm, __shfl_xor(tm, 2, 32));
      tm = fmaxf(tm, __shfl_xor(tm, 4, 32));
      tm = fmaxf(tm, __shfl_xor(tm, 8, 32));
      const float nm = fmaxf(m_run[g], tm);
      float e = expf(c[g] - nm);
      e += __shfl_xor(e, 1, 32);
      e += __shfl_xor(e, 2, 32);
      e += __shfl_xor(e, 4, 32);
      e += __shfl_xor(e, 8, 32);
      l_run[g] = l_run[g] * expf(m_run[g] - nm) + e;
      m_run[g] = nm;
    }
  }

  // ---- combine per-wave stats ----------------------------------------------
  if (ln == 0) {  // lanes 0 and 16: rows 0-7 and 8-15 respectively
#pragma unroll
    for (int g = 0; g < 8; ++g) {
      wstat[wv][g + 8 * lh][0] = m_run[g];
      wstat[wv][g + 8 * lh][1] = l_run[g];
    }
  }
  __syncthreads();
  if (tid < 16) {
    float M = -1e30f;
    for (int w = 0; w < 8; ++w) M = fmaxf(M, wstat[w][tid][0]);
    float L = 0.f;
    for (int w = 0; w < 8; ++w) L += wstat[w][tid][1] * expf(wstat[w][tid][0] - M);
    rowstat[tid][0] = M;
    rowstat[tid][1] = 1.0f / L;
  }
  __syncthreads();

  float Mv[8], Li[8];
#pragma unroll
  for (int g = 0; g < 8; ++g) {
    Mv[g] = rowstat[g + 8 * lh][0];
    Li[g] = rowstat[g + 8 * lh][1];
  }

  // ---- phase 2: normalized probs -> attn output + P·V via WMMA -------------
  v8f cv[4];
#pragma unroll
  for (int i = 0; i < 4; ++i) cv[i] = (v8f){0.f, 0.f, 0.f, 0.f, 0.f, 0.f, 0.f, 0.f};

  float* attnbase = attn + (bh * 2048 + q0) * 2048;
  const float* vbase = V + bh * 2048 * 64;

#pragma unroll
  for (int t = 0; t < 16; ++t) {
    const int n0 = (t * 8 + wv) * 16;      // key positions of this P tile

    // preload the 16 V fragments (global) for this key block
    v2f vfr[16];                           // [nv][k4] flattened
#pragma unroll
    for (int nv = 0; nv < 4; ++nv)
#pragma unroll
      for (int k4 = 0; k4 < 4; ++k4) {
        const float* vp = vbase + (size_t)(n0 + k4 * 4 + 2 * lh) * 64 + nv * 16 + ln;
        vfr[nv * 4 + k4].x = vp[0];
        vfr[nv * 4 + k4].y = vp[64];
      }

    // normalized probs: write to attn output + per-wave LDS transpose stage
#pragma unroll
    for (int g = 0; g < 8; ++g) {
      const float pv = expf(ct[t][g] - Mv[g]) * Li[g];
      attnbase[(size_t)(g + 8 * lh) * 2048 + n0 + ln] = pv;
      pstage[wv][g + 8 * lh][ln] = pv;
    }
#pragma unroll
    for (int nv = 0; nv < 4; ++nv) {
#pragma unroll
      for (int k4 = 0; k4 < 4; ++k4) {
        v2f a;
        a.x = pstage[wv][ln][k4 * 4 + 2 * lh];
        a.y = pstage[wv][ln][k4 * 4 + 2 * lh + 1];
        cv[nv] = WMMA_F32(a, vfr[nv * 4 + k4], cv[nv]);
      }
    }
  }

  // ---- reduce 8 per-wave ctx partials and store ----------------------------
#pragma unroll
  for (int nv = 0; nv < 4; ++nv)
#pragma unroll
    for (int g = 0; g < 8; ++g)
      ctxpart[wv][g + 8 * lh][nv * 16 + ln] = cv[nv][g];
  __syncthreads();
#pragma unroll
  for (int i = 0; i < 4; ++i) {
    const int idx = tid + i * 256;         // 16*64 = 1024 elements
    const int r = idx >> 6, d = idx & 63;
    float sum = 0.f;
#pragma unroll
    for (int w = 0; w < 8; ++w) sum += ctxpart[w][r][d];
    ctx[((size_t)b * 2048 + q0 + r) * 768 + h * 64 + d] = sum;
  }
}

// ---------------------------------------------------------------------------
extern "C" void kernel_launch(void* const* d_in, const int* in_sizes, int n_in,
                              void* d_out, int out_size, void* d_ws, size_t ws_size,
                              hipStream_t stream) {
  (void)in_sizes; (void)n_in; (void)out_size; (void)ws_size;
  const float* q    = (const float*)d_in[0];
  const float* k    = (const float*)d_in[1];
  const float* v    = (const float*)d_in[2];
  const float* mask = (const float*)d_in[3];
  const float* Wq   = (const float*)d_in[4];
  const float* bq   = (const float*)d_in[5];
  const float* Wk   = (const float*)d_in[6];
  const float* bk   = (const float*)d_in[7];
  const float* Wv   = (const float*)d_in[8];
  const float* bv   = (const float*)d_in[9];
  const float* Wo   = (const float*)d_in[10];
  const float* bo   = (const float*)d_in[11];

  float* out  = (float*)d_out;
  float* attn = out + (size_t)2 * 2048 * 768;   // outputs: (out, attn) concatenated

  const size_t per = (size_t)2 * 12 * 2048 * 64;  // 3,145,728 floats
  float* Qbuf  = (float*)d_ws;
  float* KTbuf = Qbuf + per;
  float* Vbuf  = KTbuf + per;
  float* Cbuf  = Vbuf + per;

  const dim3 gg(768 / 64, 4096 / 64);   // (12, 64) tiles
  gemm_bias_64x64<<<gg, 256, 0, stream>>>(q, Wq, bq, Qbuf, 4096, 768, 768, 1);
  gemm_bias_64x64<<<gg, 256, 0, stream>>>(k, Wk, bk, KTbuf, 4096, 768, 768, 2);
  gemm_bias_64x64<<<gg, 256, 0, stream>>>(v, Wv, bv, Vbuf, 4096, 768, 768, 1);

  attention_kernel<<<dim3(2 * 12 * 128), 256, 0, stream>>>(Qbuf, KTbuf, Vbuf, mask, attn, Cbuf);

  gemm_bias_64x64<<<gg, 256, 0, stream>>>(Cbuf, Wo, bo, out, 4096, 768, 768, 0);
}